// SparsityAwareRouter_82884278878586
// MI455X (gfx1250) — compile-verified
//
#include <hip/hip_runtime.h>
#include <hip/hip_bf16.h>
#include <math.h>

typedef __attribute__((ext_vector_type(2))) float v2f;
typedef __attribute__((ext_vector_type(8))) float v8f;

// ---------------------------------------------------------------------------
// Stage 0: bias_mean = mean(bias)  (F elements, one block)
// ---------------------------------------------------------------------------
__global__ void sar_biasmean_kernel(const float* __restrict__ bias,
                                    float* __restrict__ out, int F) {
    __shared__ float red[256];
    float s = 0.f;
    for (int i = threadIdx.x; i < F; i += blockDim.x) s += bias[i];
    red[threadIdx.x] = s;
    __syncthreads();
    for (int off = 128; off > 0; off >>= 1) {
        if ((int)threadIdx.x < off) red[threadIdx.x] += red[threadIdx.x + off];
        __syncthreads();
    }
    if (threadIdx.x == 0) out[0] = red[0] / (float)F;
}

// ---------------------------------------------------------------------------
// Stage 1: per-column stats of W_base / W_delta over F rows.
//   mb[d], md[d], var_b[d], cov_bd[d], var_d[d]
// Written into two 16-column matrices V1 (mu side) / V2 (var side) in the
// WMMA-B interleaved layout:  V[(d>>1)*32 + col*2 + (d&1)]
//   V1: col0 = mb, col1 = md, cols 2..15 = 0
//   V2: col0 = var_b, col1 = cov_bd, col2 = var_d, cols 3..15 = 0
// ---------------------------------------------------------------------------
__global__ void sar_colstats_kernel(const float* __restrict__ Wb,
                                    const float* __restrict__ Wd,
                                    float* __restrict__ V1,
                                    float* __restrict__ V2,
                                    int F, int D) {
    int d = blockIdx.x * blockDim.x + threadIdx.x;
    if (d >= D) return;
    float smb = 0.f, smd = 0.f, sbb = 0.f, sdd = 0.f, sbd = 0.f;
    const float* pb = Wb + d;
    const float* pd = Wd + d;
    for (int f = 0; f < F; ++f) {
        float b = pb[(size_t)f * D];
        float w = pd[(size_t)f * D];
        smb += b; smd += w;
        sbb = fmaf(b, b, sbb);
        sdd = fmaf(w, w, sdd);
        sbd = fmaf(b, w, sbd);
    }
    float inv = 1.0f / (float)F;
    float mb = smb * inv, md = smd * inv;
    float varb  = sbb * inv - mb * mb;
    float covbd = sbd * inv - mb * md;
    float vard  = sdd * inv - md * md;

    size_t base = (size_t)(d >> 1) * 32 + (size_t)(d & 1);
    #pragma unroll
    for (int c = 0; c < 16; ++c) { V1[base + c * 2] = 0.f; V2[base + c * 2] = 0.f; }
    V1[base + 0] = mb;    V1[base + 2] = md;
    V2[base + 0] = varb;  V2[base + 2] = covbd;  V2[base + 4] = vard;
}

// ---------------------------------------------------------------------------
// Stage 2: per-token router.  One wave32 owns 16 tokens.
//   acc1 = X  (16xD) * V1 (Dx16)   -> p1, p2 in cols 0,1
//   acc2 = X^2(16xD) * V2 (Dx16)   -> q1, q2, q3 in cols 0,1,2
// via V_WMMA_F32_16X16X4_F32 over K = D in steps of 4 (full fp32 accum).
// Epilogue: erf logits for E experts, top-2 softmax gating.
// ---------------------------------------------------------------------------
__global__ __launch_bounds__(256) void sar_router_kernel(
    const float* __restrict__ x,
    const float* __restrict__ V1,
    const float* __restrict__ V2,
    const float* __restrict__ alpha,
    const float* __restrict__ beta,
    const float* __restrict__ biasmean,
    float* __restrict__ out_w,
    float* __restrict__ out_l,
    int D, int E) {
    __shared__ float lds[8 * 544];               // 8 waves * (2 tiles * 16*17)

    const int lane = threadIdx.x & 31;
    const int wslot = threadIdx.x >> 5;
    const int gwave = blockIdx.x * 8 + wslot;
    const int tok_base = gwave * 16;
    const int half = lane >> 4;                  // 0: K {0,1}, 1: K {2,3}
    const int l15 = lane & 15;

    // A fragment source: row = token l15, K offset = 2*half  (ISA 16x4 f32 map)
    const float* px  = x + (size_t)(tok_base + l15) * D + half * 2;
    // B fragment source: col = l15, rows K / K+1 interleaved contiguously
    const float* pb1 = V1 + l15 * 2 + half * 32;
    const float* pb2 = V2 + l15 * 2 + half * 32;

    v8f acc1 = {};
    v8f acc2 = {};
    for (int k0 = 0; k0 < D; k0 += 4) {
        v2f a  = *(const v2f*)(px + k0);
        v2f a2 = a * a;
        size_t boff = (size_t)(k0 >> 1) * 32;
        v2f b1 = *(const v2f*)(pb1 + boff);
        v2f b2 = *(const v2f*)(pb2 + boff);
        acc1 = __builtin_amdgcn_wmma_f32_16x16x4_f32(false, a,  false, b1,
                                                     (short)0, acc1, false, false);
        acc2 = __builtin_amdgcn_wmma_f32_16x16x4_f32(false, a2, false, b2,
                                                     (short)0, acc2, false, false);
    }

    // Spill C tiles to LDS (C map: lane holds col l15, rows j + 8*half).
    float* slot = lds + wslot * 544;
    #pragma unroll
    for (int j = 0; j < 8; ++j) {
        int row = j + half * 8;
        slot[row * 17 + l15]       = acc1[j];
        slot[272 + row * 17 + l15] = acc2[j];
    }
    __syncthreads();

    if (lane < 16) {
        const int token = tok_base + lane;
        float p1 = slot[lane * 17 + 0];
        float p2 = slot[lane * 17 + 1];
        float q1 = slot[272 + lane * 17 + 0];
        float q2 = slot[272 + lane * 17 + 1];
        float q3 = slot[272 + lane * 17 + 2];
        const float bm = biasmean[0];
        const float inv_sqrt2 = 0.70710678118654752f;

        float lg[8];
        float v1 = -2.0f, v2 = -2.0f;            // erf in (-1,1)
        int   i1 = 0,     i2 = 0;
        for (int e = 0; e < E; ++e) {
            float ae = alpha[e], be = beta[e];
            float mu = fmaf(ae, p1, fmaf(be, p2, bm));
            float s2 = ae * ae * q1 + 2.0f * ae * be * q2 + be * be * q3;
            float sig = sqrtf(s2 + 1e-8f);
            float z = (mu / sig) * inv_sqrt2;
            float L = erff(z);
            lg[e] = L;
            if (L > v1)      { v2 = v1; i2 = i1; v1 = L; i1 = e; }
            else if (L > v2) { v2 = L; i2 = e; }
        }
        // softmax over the two top logits (v1 >= v2)
        float e2 = __expf(v2 - v1);
        float w1 = 1.0f / (1.0f + e2);
        float w2 = e2 * w1;

        float wgt[8];
        #pragma unroll
        for (int e = 0; e < 8; ++e) wgt[e] = 0.f;
        wgt[i1] = w1;
        wgt[i2] = w2;

        float4* ow = (float4*)(out_w + (size_t)token * 8);
        float4* ol = (float4*)(out_l + (size_t)token * 8);
        ow[0] = make_float4(wgt[0], wgt[1], wgt[2], wgt[3]);
        ow[1] = make_float4(wgt[4], wgt[5], wgt[6], wgt[7]);
        ol[0] = make_float4(lg[0], lg[1], lg[2], lg[3]);
        ol[1] = make_float4(lg[4], lg[5], lg[6], lg[7]);
    }
}

// ---------------------------------------------------------------------------
extern "C" void kernel_launch(void* const* d_in, const int* in_sizes, int n_in,
                              void* d_out, int out_size, void* d_ws, size_t ws_size,
                              hipStream_t stream) {
    const float* x     = (const float*)d_in[0];
    const float* Wb    = (const float*)d_in[1];
    const float* Wd    = (const float*)d_in[2];
    const float* bias  = (const float*)d_in[3];
    const float* alpha = (const float*)d_in[4];
    const float* beta  = (const float*)d_in[5];

    const int F = in_sizes[3];                 // 4096
    const int E = in_sizes[4];                 // 8
    const int D = in_sizes[1] / F;             // 1024
    const int tokens = in_sizes[0] / D;        // 16384 (B*S)

    float* ws_f     = (float*)d_ws;
    float* biasmean = ws_f;                    // 1 float (padded to 16)
    float* V1       = ws_f + 16;               // D*16 floats
    float* V2       = V1 + (size_t)D * 16;     // D*16 floats

    float* out_w = (float*)d_out;              // router_weights [tokens, E]
    float* out_l = out_w + (size_t)tokens * E; // router_logits  [tokens, E]

    sar_biasmean_kernel<<<1, 256, 0, stream>>>(bias, biasmean, F);
    sar_colstats_kernel<<<(D + 255) / 256, 256, 0, stream>>>(Wb, Wd, V1, V2, F, D);

    const int waves  = tokens / 16;            // 1024
    const int blocks = waves / 8;              // 128 blocks x 256 threads
    sar_router_kernel<<<blocks, 256, 0, stream>>>(x, V1, V2, alpha, beta,
                                                  biasmean, out_w, out_l, D, E);
}